// SpGraphAttentionLayer_42614665511372
// MI455X (gfx1250) — compile-verified
//
#include <hip/hip_runtime.h>
#include <math.h>

#define N_NODES 50000
#define N_EDGES 1600000
#define IN_F    256
#define OUT_F   128
#define ALPHA   0.2f
#define EPS     1e-9f
#define ROWS    32            // rows (M) per block: two 16-row WMMA tiles

#ifndef USE_TDM
#define USE_TDM 1             // Tensor Data Mover staging of the A tile
#endif

typedef __attribute__((ext_vector_type(2))) float    v2f;
typedef __attribute__((ext_vector_type(8))) float    v8f;
typedef __attribute__((ext_vector_type(4))) unsigned v4u;
typedef __attribute__((ext_vector_type(8))) unsigned v8u;

// -------- Kernel 1: h = x @ W using V_WMMA_F32_16X16X4_F32 ------------------
// 256 threads = 8 waves; wave w owns N-tile w (cols 16w..16w+15) and computes
// TWO 16x16 output tiles (rows m0..m0+15 and m0+16..m0+31) so each B fetch
// feeds two WMMAs. A tile staged into LDS by the TDM (padded stride 260).
__global__ __launch_bounds__(256) void gat_gemm_wmma(
    const float* __restrict__ x, const float* __restrict__ W,
    float* __restrict__ h)
{
    __shared__ float xs[ROWS * 260];          // 32x256 tile, 260-float stride
    const int tid = threadIdx.x;
    const int m0  = blockIdx.x * ROWS;

#if USE_TDM
    if (tid < 32) {                           // wave 0 issues one TDM descriptor
        const unsigned long long ga =
            (unsigned long long)(const void*)(x + (size_t)m0 * IN_F);
        const unsigned lds = (unsigned)(unsigned long long)(void*)xs;
        const unsigned rows_left = (unsigned)(N_NODES - m0);   // OOB rows -> 0

        v4u g0;
        g0[0] = 1u;                                         // count=1 (valid)
        g0[1] = lds;                                        // lds_addr
        g0[2] = (unsigned)(ga & 0xffffffffu);               // global_addr[31:0]
        g0[3] = (unsigned)((ga >> 32) & 0x01ffffffu)        // global_addr[56:32]
              | (2u << 30);                                 // type=2 ("image")

        v8u g1;
        g1[0] = (2u << 16)                                  // data_size = 4B
              | (1u << 20)                                  // pad_enable
              | (7u << 22)                                  // pad_interval: 256 DW
              | (3u << 25);                                 // pad_amount: 4 DW
        g1[1] = ((unsigned)IN_F & 0xffffu) << 16;           // tensor_dim0 lo16
        g1[2] = (rows_left & 0xffffu) << 16;                // dim0 hi=0 | dim1 lo16
        g1[3] = ((rows_left >> 16) & 0xffffu)               // tensor_dim1 hi16
              | ((unsigned)IN_F << 16);                     // tile_dim0 = 256
        g1[4] = (unsigned)ROWS;                             // tile_dim1=32, tile_dim2=0
        g1[5] = (unsigned)IN_F;                             // tensor_dim0_stride lo32
        g1[6] = 0u;                                         // stride hi | dim1_stride lo
        g1[7] = 0u;

        asm volatile("tensor_load_to_lds %0, %1" :: "s"(g0), "s"(g1) : "memory");
        __builtin_amdgcn_s_wait_tensorcnt(0);
    }
    __syncthreads();
#else
    for (int i = tid; i < ROWS * (IN_F / 4); i += 256) {
        const int m  = i >> 6;
        const int k4 = (i & 63) * 4;
        const int gm = (m0 + m < N_NODES) ? (m0 + m) : (N_NODES - 1);
        const float4 v = *(const float4*)(x + (size_t)gm * IN_F + k4);
        float* d = xs + m * 260 + k4;
        d[0] = v.x; d[1] = v.y; d[2] = v.z; d[3] = v.w;
    }
    __syncthreads();
#endif

    const int lane = tid & 31;
    const int wave = tid >> 5;
    const int hi   = lane >> 4;           // half-wave select
    const int ml   = lane & 15;           // A row / B,C,D column lane
    const int n    = wave * 16 + ml;

    v8f acc0 = {}, acc1 = {};
    #pragma unroll 4
    for (int k0 = 0; k0 < IN_F; k0 += 4) {
        const int k = k0 + hi * 2;
        v2f B, A0, A1;
        B[0]  = W[(size_t)k * OUT_F + n];
        B[1]  = W[(size_t)(k + 1) * OUT_F + n];
        A0[0] = xs[ml * 260 + k];
        A0[1] = xs[ml * 260 + k + 1];
        A1[0] = xs[(16 + ml) * 260 + k];
        A1[1] = xs[(16 + ml) * 260 + k + 1];
        acc0 = __builtin_amdgcn_wmma_f32_16x16x4_f32(
            false, A0, false, B, (short)0, acc0, false, false);
        acc1 = __builtin_amdgcn_wmma_f32_16x16x4_f32(
            false, A1, false, B, (short)0, acc1, false, false);
    }

    #pragma unroll
    for (int v = 0; v < 8; ++v) {
        h[(size_t)(m0 + v + 8 * hi) * OUT_F + n] = acc0[v];
        const int r1 = m0 + 16 + v + 8 * hi;
        if (r1 < N_NODES)
            h[(size_t)r1 * OUT_F + n] = acc1[v];
    }
}

// -------- Kernel 2: per-node scores s_src = h.a[:F], s_dst = h.a[F:] --------
__global__ __launch_bounds__(256) void gat_scores(
    const float* __restrict__ h, const float* __restrict__ a,
    float* __restrict__ s_src, float* __restrict__ s_dst)
{
    const int lane = threadIdx.x & 31;
    const int wave = threadIdx.x >> 5;
    const int node = blockIdx.x * 8 + wave;
    if (node >= N_NODES) return;

    const float4 hv = *(const float4*)(h + (size_t)node * OUT_F + lane * 4);
    const float4 as = *(const float4*)(a + lane * 4);
    const float4 ad = *(const float4*)(a + OUT_F + lane * 4);
    float s1 = hv.x * as.x + hv.y * as.y + hv.z * as.z + hv.w * as.w;
    float s2 = hv.x * ad.x + hv.y * ad.y + hv.z * ad.z + hv.w * ad.w;

    #pragma unroll
    for (int off = 16; off > 0; off >>= 1) {
        s1 += __shfl_xor(s1, off, 32);
        s2 += __shfl_xor(s2, off, 32);
    }
    if (lane == 0) { s_src[node] = s1; s_dst[node] = s2; }
}

// -------- Kernel 3: edge scatter (L2-resident: h & agg both fit in 192MB) ---
__global__ __launch_bounds__(256) void gat_edges(
    const int* __restrict__ src_idx, const int* __restrict__ dst_idx,
    const float* __restrict__ h,
    const float* __restrict__ s_src, const float* __restrict__ s_dst,
    float* __restrict__ agg, float* __restrict__ rowsum)
{
    const int lane = threadIdx.x & 31;
    const int wave = threadIdx.x >> 5;
    const long long e = (long long)blockIdx.x * 8 + wave;
    if (e >= N_EDGES) return;

    const int s = src_idx[e];
    const int d = dst_idx[e];

    const float score = s_src[s] + s_dst[d];
    const float lr = score > 0.f ? score : ALPHA * score;
    const float w  = expf(-lr);

    if (lane == 0) atomicAdd(rowsum + s, w);

    const float4 hv = *(const float4*)(h + (size_t)d * OUT_F + lane * 4);
    float* ap = agg + (size_t)s * OUT_F + lane * 4;
    atomicAdd(ap + 0, w * hv.x);
    atomicAdd(ap + 1, w * hv.y);
    atomicAdd(ap + 2, w * hv.z);
    atomicAdd(ap + 3, w * hv.w);
}

// -------- Kernel 4: h' = agg / (rowsum + eps); out = elu(h') ----------------
__global__ __launch_bounds__(256) void gat_finalize(
    float* __restrict__ out, const float* __restrict__ rowsum)
{
    const size_t i = (size_t)blockIdx.x * 256 + threadIdx.x;
    if (i >= (size_t)N_NODES * OUT_F) return;
    const int node = (int)(i / OUT_F);
    const float v = out[i] / (rowsum[node] + EPS);
    out[i] = v > 0.f ? v : (expf(v) - 1.f);
}

extern "C" void kernel_launch(void* const* d_in, const int* in_sizes, int n_in,
                              void* d_out, int out_size, void* d_ws, size_t ws_size,
                              hipStream_t stream)
{
    const float* x     = (const float*)d_in[0];
    const int*   edges = (const int*)d_in[1];   // [2, N_EDGES]: src then dst
    const float* W     = (const float*)d_in[2];
    const float* a     = (const float*)d_in[3];
    float*       out   = (float*)d_out;         // used as agg, then finalized

    float* h      = (float*)d_ws;                         // N_NODES*OUT_F
    float* s_src  = h + (size_t)N_NODES * OUT_F;          // N_NODES
    float* s_dst  = s_src + N_NODES;                      // N_NODES
    float* rowsum = s_dst + N_NODES;                      // N_NODES

    hipMemsetAsync(out,    0, (size_t)N_NODES * OUT_F * sizeof(float), stream);
    hipMemsetAsync(rowsum, 0, (size_t)N_NODES * sizeof(float), stream);

    gat_gemm_wmma<<<(N_NODES + ROWS - 1) / ROWS, 256, 0, stream>>>(x, W, h);
    gat_scores  <<<(N_NODES + 7) / 8, 256, 0, stream>>>(h, a, s_src, s_dst);
    gat_edges   <<<(N_EDGES + 7) / 8, 256, 0, stream>>>(
        edges, edges + N_EDGES, h, s_src, s_dst, out, rowsum);
    gat_finalize<<<(int)(((size_t)N_NODES * OUT_F + 255) / 256), 256, 0, stream>>>(
        out, rowsum);
}